// MultiHeadedAttentionHighDim_55482387530265
// MI455X (gfx1250) — compile-verified
//
#include <hip/hip_runtime.h>
#include <hip/hip_bf16.h>
#include <math.h>

// ---------------------------------------------------------------------------
// MI455X (gfx1250) implementation, fp32 end-to-end via V_WMMA_F32_16X16X4_F32.
// Problem is HBM-bound (~113 GFLOP vs ~0.3-0.7 GB traffic @ 23.3 TB/s), so we
// keep fp32 precision and use the f32 WMMA path instead of down-converting.
// ---------------------------------------------------------------------------

typedef __attribute__((ext_vector_type(2))) float v2f;
typedef __attribute__((ext_vector_type(8))) float v8f;

#define WMMA_F32(a, b, c) \
    __builtin_amdgcn_wmma_f32_16x16x4_f32(false, (a), false, (b), (short)0, (c), false, false)

static __device__ __forceinline__ v2f ld2(const float* p) {
    return *reinterpret_cast<const v2f*>(p);
}

// ---------------------------------------------------------------------------
// Y[m][n] = sum_k X[m][k] * W[n][k] + bias[n]
// X: [M x 512], W: [512 x 512] row-major, Y: [M x 512]. M multiple of 128.
// grid = (M/128, 512/64), block = 256 threads (8 waves).
// Wave w computes rows [blk.x*128 + 16w, +16), cols [blk.y*64, +64) as 4 WMMA
// tiles. A/B per-lane layout (ISA 7.12.2): row = lane%16, K = 2*(lane/16)+vgpr.
// ---------------------------------------------------------------------------
__global__ __launch_bounds__(256)
void proj_gemm(const float* __restrict__ X, const float* __restrict__ W,
               const float* __restrict__ bias, float* __restrict__ Y) {
    const int lane = threadIdx.x & 31;
    const int wave = threadIdx.x >> 5;
    const int lm   = lane & 15;
    const int kg   = (lane >> 4) << 1;        // K offset: 0 or 2
    const int m0   = blockIdx.x * 128 + wave * 16;
    const int n0   = blockIdx.y * 64;

    v8f c0 = {}, c1 = {}, c2 = {}, c3 = {};
    const float* xrow = X + (size_t)(m0 + lm) * 512;
    const float* wrow = W + (size_t)(n0 + lm) * 512;

#pragma unroll 4
    for (int k0 = 0; k0 < 512; k0 += 4) {
        v2f a  = ld2(xrow + k0 + kg);
        v2f b0 = ld2(wrow + k0 + kg);
        v2f b1 = ld2(wrow + 16 * 512 + k0 + kg);
        v2f b2 = ld2(wrow + 32 * 512 + k0 + kg);
        v2f b3 = ld2(wrow + 48 * 512 + k0 + kg);
        c0 = WMMA_F32(a, b0, c0);
        c1 = WMMA_F32(a, b1, c1);
        c2 = WMMA_F32(a, b2, c2);
        c3 = WMMA_F32(a, b3, c3);
    }

    const float bn0 = bias[n0 + lm];
    const float bn1 = bias[n0 + 16 + lm];
    const float bn2 = bias[n0 + 32 + lm];
    const float bn3 = bias[n0 + 48 + lm];
#pragma unroll
    for (int r = 0; r < 8; ++r) {
        const int m = m0 + r + 4 * kg;        // C/D layout: M = r + 8*(lane/16)
        float* y = Y + (size_t)m * 512 + n0 + lm;
        y[0]  = c0[r] + bn0;
        y[16] = c1[r] + bn1;
        y[32] = c2[r] + bn2;
        y[48] = c3[r] + bn3;
    }
}

// ---------------------------------------------------------------------------
// Per-(b,s,h) attention over time: scores=QK^T/8, mask, softmax, x=P@V.
// One workgroup per (b,s,h): grid = 2048, block = 256 (8 waves).
// x overwrites V's buffer in place (disjoint head slices per workgroup;
// __syncthreads separates last V read from store).
// ---------------------------------------------------------------------------
__global__ __launch_bounds__(256)
void attn_kernel(const float* __restrict__ Qp, const float* __restrict__ Kp,
                 float* __restrict__ Vp, const unsigned char* __restrict__ mask) {
    __shared__ float sc[128][128];            // 64 KB score/prob tile
    const int lane = threadIdx.x & 31;
    const int wave = threadIdx.x >> 5;
    const int lm   = lane & 15;
    const int kg   = (lane >> 4) << 1;
    const int h = blockIdx.x & 7;
    const int s = (blockIdx.x >> 3) & 63;
    const int b = blockIdx.x >> 9;

    const size_t tok = (size_t)(b * 64 + s) * 128;
    const float* Q = Qp + tok * 512 + h * 64;
    const float* K = Kp + tok * 512 + h * 64;
    float*       V = Vp + tok * 512 + h * 64;
    const unsigned char* msk = mask + (size_t)(b * 64 + s) * 128 * 128;

    const int t0 = wave * 16;                 // this wave's score rows

    // ---- stage 1: scores[t0..t0+15][*] = Q K^T / sqrt(64), masked ----
    const float* qrow = Q + (size_t)(t0 + lm) * 512;
#pragma unroll
    for (int ut = 0; ut < 8; ++ut) {
        const int u0 = ut * 16;
        v8f c = {};
        const float* krow = K + (size_t)(u0 + lm) * 512;
#pragma unroll
        for (int k0 = 0; k0 < 64; k0 += 4) {
            v2f a  = ld2(qrow + k0 + kg);
            v2f bb = ld2(krow + k0 + kg);
            c = WMMA_F32(a, bb, c);
        }
#pragma unroll
        for (int r = 0; r < 8; ++r) {
            const int t = t0 + r + 4 * kg;
            const int u = u0 + lm;
            sc[t][u] = msk[t * 128 + u] ? c[r] * 0.125f : -1e9f;
        }
    }
    __syncthreads();

    // ---- stage 2: row softmax (wave handles rows t0..t0+15) ----
    for (int rr = 0; rr < 16; ++rr) {
        const int t = t0 + rr;
        float v0 = sc[t][lane], v1 = sc[t][lane + 32];
        float v2 = sc[t][lane + 64], v3 = sc[t][lane + 96];
        float m = fmaxf(fmaxf(v0, v1), fmaxf(v2, v3));
#pragma unroll
        for (int off = 16; off >= 1; off >>= 1)
            m = fmaxf(m, __shfl_xor(m, off, 32));
        v0 = __expf(v0 - m); v1 = __expf(v1 - m);
        v2 = __expf(v2 - m); v3 = __expf(v3 - m);
        float sum = v0 + v1 + v2 + v3;
#pragma unroll
        for (int off = 16; off >= 1; off >>= 1)
            sum += __shfl_xor(sum, off, 32);
        const float inv = 1.0f / sum;
        sc[t][lane]      = v0 * inv;
        sc[t][lane + 32] = v1 * inv;
        sc[t][lane + 64] = v2 * inv;
        sc[t][lane + 96] = v3 * inv;
    }
    __syncthreads();

    // ---- stage 3: x[t0..t0+15][0..63] = P @ V ----
    v8f c0 = {}, c1 = {}, c2 = {}, c3 = {};
    for (int k0 = 0; k0 < 128; k0 += 4) {
        const int ka = k0 + kg;
        v2f a = ld2(&sc[t0 + lm][ka]);        // A from LDS
        const float* vr0 = V + (size_t)ka * 512;
        const float* vr1 = vr0 + 512;
        v2f b0 = { vr0[lm],      vr1[lm]      };
        v2f b1 = { vr0[16 + lm], vr1[16 + lm] };
        v2f b2 = { vr0[32 + lm], vr1[32 + lm] };
        v2f b3 = { vr0[48 + lm], vr1[48 + lm] };
        c0 = WMMA_F32(a, b0, c0);
        c1 = WMMA_F32(a, b1, c1);
        c2 = WMMA_F32(a, b2, c2);
        c3 = WMMA_F32(a, b3, c3);
    }
    __syncthreads();                          // all V reads complete before overwrite
#pragma unroll
    for (int r = 0; r < 8; ++r) {
        const int t = t0 + r + 4 * kg;
        float* xr = V + (size_t)t * 512;
        xr[lm]      = c0[r];
        xr[16 + lm] = c1[r];
        xr[32 + lm] = c2[r];
        xr[48 + lm] = c3[r];
    }
}

// ---------------------------------------------------------------------------
// Sensor Gram: Ssens[b][i][j] += sum_{f in chunk} Qs[b,i,f] * Ks[b,j,f]
// grid = (4 batches, 32 F-chunks of 2048), block = 256 (8 waves, 2 tiles each).
// Split-K partials accumulated with atomicAdd into a zeroed buffer.
// ---------------------------------------------------------------------------
__global__ __launch_bounds__(256)
void sensor_gram(const float* __restrict__ Qs, const float* __restrict__ Ks,
                 float* __restrict__ Ssens) {
    const int lane = threadIdx.x & 31;
    const int wave = threadIdx.x >> 5;
    const int lm   = lane & 15;
    const int kg   = (lane >> 4) << 1;
    const int b     = blockIdx.x;
    const int chunk = blockIdx.y;
    const size_t F  = 65536;
    const size_t f0 = (size_t)chunk * 2048;
    const float* Qb = Qs + (size_t)b * 64 * F;
    const float* Kb = Ks + (size_t)b * 64 * F;

    for (int p = wave * 2; p < wave * 2 + 2; ++p) {
        const int it = p >> 2, jt = p & 3;
        v8f c = {};
        const float* qrow = Qb + (size_t)(it * 16 + lm) * F + f0;
        const float* krow = Kb + (size_t)(jt * 16 + lm) * F + f0;
#pragma unroll 4
        for (int k0 = 0; k0 < 2048; k0 += 4) {
            v2f a  = ld2(qrow + k0 + kg);
            v2f bb = ld2(krow + k0 + kg);
            c = WMMA_F32(a, bb, c);
        }
#pragma unroll
        for (int r = 0; r < 8; ++r) {
            const int i = it * 16 + r + 4 * kg;
            const int j = jt * 16 + lm;
            atomicAdd(&Ssens[((size_t)b * 64 + i) * 64 + j], c[r]);
        }
    }
}

// ---------------------------------------------------------------------------
// Beta-softmax over sensor axis: rows = B*S = 256, length 64.
// scale = 1/sqrt(T*D) = 1/256.  out = exp((x-max)*beta) / sum(...)
// ---------------------------------------------------------------------------
__global__ __launch_bounds__(256)
void sensor_softmax(const float* __restrict__ Ssens, const float* __restrict__ beta_p,
                    float* __restrict__ out) {
    const int r = blockIdx.x * blockDim.x + threadIdx.x;
    if (r >= 256) return;
    const float beta  = beta_p[0];
    const float scale = 1.0f / 256.0f;
    const float* row = Ssens + (size_t)r * 64;
    float m = -1e30f;
    for (int j = 0; j < 64; ++j) m = fmaxf(m, row[j] * scale);
    float sum = 0.0f;
    for (int j = 0; j < 64; ++j) sum += __expf((row[j] * scale - m) * beta);
    const float inv = 1.0f / sum;
    float* o = out + (size_t)r * 64;
    for (int j = 0; j < 64; ++j) o[j] = __expf((row[j] * scale - m) * beta) * inv;
}

// ---------------------------------------------------------------------------
// Host launcher. Workspace: 3 x 16.78M floats (201 MB) + 16K floats.
// Input order: query,key,value,mask,beta, then (W,b) for q,k,v,o,qs,ks,vs.
// Output: out [B,S,T,D] (16,777,216 f32) then attn_s [B,1,S,S] (16,384 f32).
// ---------------------------------------------------------------------------
extern "C" void kernel_launch(void* const* d_in, const int* in_sizes, int n_in,
                              void* d_out, int out_size, void* d_ws, size_t ws_size,
                              hipStream_t stream) {
    const float* query = (const float*)d_in[0];
    const float* key   = (const float*)d_in[1];
    const float* value = (const float*)d_in[2];
    const unsigned char* mask = (const unsigned char*)d_in[3];
    const float* beta = (const float*)d_in[4];
    const float* Wq  = (const float*)d_in[5];  const float* bq  = (const float*)d_in[6];
    const float* Wk  = (const float*)d_in[7];  const float* bk  = (const float*)d_in[8];
    const float* Wv  = (const float*)d_in[9];  const float* bv  = (const float*)d_in[10];
    const float* Wo  = (const float*)d_in[11]; const float* bo  = (const float*)d_in[12];
    const float* Wqs = (const float*)d_in[13]; const float* bqs = (const float*)d_in[14];
    const float* Wks = (const float*)d_in[15]; const float* bks = (const float*)d_in[16];
    // d_in[17], d_in[18] (Wvs, bvs) are unused by the reference.

    float* out = (float*)d_out;
    const size_t BIG = (size_t)32768 * 512;   // 16,777,216 floats per tensor
    float* buf0 = (float*)d_ws;               // Qs -> Qp
    float* buf1 = buf0 + BIG;                 // Ks -> Kp
    float* buf2 = buf1 + BIG;                 // Vp -> x (in-place)
    float* ssen = buf2 + BIG;                 // 16,384-float sensor scores

    const dim3 blk(256);
    const dim3 gProj(256, 8);                 // (32768/128, 512/64)

    hipMemsetAsync(ssen, 0, 16384 * sizeof(float), stream);

    // Sensor branch projections + Gram (buffers reused afterwards).
    proj_gemm<<<gProj, blk, 0, stream>>>(query, Wqs, bqs, buf0);
    proj_gemm<<<gProj, blk, 0, stream>>>(key,   Wks, bks, buf1);
    sensor_gram<<<dim3(4, 32), blk, 0, stream>>>(buf0, buf1, ssen);

    // Main branch: Q/K/V projections, attention (x overwrites buf2), out proj.
    proj_gemm<<<gProj, blk, 0, stream>>>(query, Wq, bq, buf0);
    proj_gemm<<<gProj, blk, 0, stream>>>(key,   Wk, bk, buf1);
    proj_gemm<<<gProj, blk, 0, stream>>>(value, Wv, bv, buf2);
    attn_kernel<<<dim3(2048), blk, 0, stream>>>(buf0, buf1, buf2, mask);
    proj_gemm<<<gProj, blk, 0, stream>>>(buf2, Wo, bo, out);

    // Sensor beta-softmax into the tail of d_out.
    sensor_softmax<<<dim3(1), blk, 0, stream>>>(ssen, beta, out + BIG);
}